// SpatialEncoder_87840671138525
// MI455X (gfx1250) — compile-verified
//
#include <hip/hip_runtime.h>
#include <hip/hip_bf16.h>
#include <math.h>

typedef float v2f __attribute__((ext_vector_type(2)));
typedef float v8f __attribute__((ext_vector_type(8)));

#define NEG_SLOPE 0.2f
#define VAR_EPS   1e-4f
#define CSP 64      // N_SPATIAL
#define KL  256     // N_LATENT

__device__ __forceinline__ float lrelu(float x) { return x > 0.0f ? x : NEG_SLOPE * x; }

// order-preserving float<->uint for atomicMax on signed floats
__device__ __forceinline__ unsigned flip_f(float f) {
    unsigned u = __float_as_uint(f);
    return (u & 0x80000000u) ? ~u : (u | 0x80000000u);
}
__device__ __forceinline__ float unflip_f(unsigned u) {
    return __uint_as_float((u & 0x80000000u) ? (u & 0x7FFFFFFFu) : ~u);
}

__device__ __forceinline__ v8f wmma_f32(v2f a, v2f b, v8f c) {
    return __builtin_amdgcn_wmma_f32_16x16x4_f32(false, a, false, b, (short)0, c, false, false);
}

// ---------------------------------------------------------------------------
// 0) init: zero spatial accumulator + softmax denom, set e_max to -inf (flipped 0)
__global__ void sp_init(float* __restrict__ spatial, float* __restrict__ denom,
                        unsigned* __restrict__ emax, int N) {
    int i = blockIdx.x * blockDim.x + threadIdx.x;
    if (i < N * CSP) spatial[i] = 0.0f;
    if (i < N) { denom[i] = 0.0f; emax[i] = 0u; }
}

// ---------------------------------------------------------------------------
// 1) fused x_l / x_r GEMM: [N,256] @ [256,64]^T twice, z streamed once (NT loads).
//    One wave = 4 row-tiles (64 rows) x 4 column tiles; per k-step 12 loads feed
//    32 WMMAs so VMEM latency is amortized.
__global__ __launch_bounds__(128)
void xlr_gemm_wmma(const float* __restrict__ z,
                   const float* __restrict__ W_l, const float* __restrict__ b_l,
                   const float* __restrict__ W_r, const float* __restrict__ b_r,
                   float* __restrict__ x_l, float* __restrict__ x_r, int N)
{
    const int lane = threadIdx.x & 31;
    const int wave = threadIdx.x >> 5;
    const int m0 = (blockIdx.x * 4 + wave) * 64;          // 4 tiles of 16 rows per wave
    if (m0 >= N) return;
    const int grp = lane >> 4;   // which K pair this lane holds
    const int lid = lane & 15;   // A row / B column within tile

    const float* zr[4];
#pragma unroll
    for (int t = 0; t < 4; ++t) {
        int r = m0 + t * 16 + lid;
        if (r >= N) r = N - 1;                            // clamp keeps EXEC full for WMMA
        zr[t] = z + (size_t)r * KL;
    }

    v8f accl[4][4] = {}, accr[4][4] = {};
    for (int k = 0; k < KL; k += 4) {
        const int k0 = k + 2 * grp;
        v2f a[4], bl[4], br[4];
#pragma unroll
        for (int t = 0; t < 4; ++t)
            a[t] = __builtin_nontemporal_load((const v2f*)(zr[t] + k0));
#pragma unroll
        for (int ct = 0; ct < 4; ++ct) {
            const int c = ct * 16 + lid;
            bl[ct] = *(const v2f*)(W_l + (size_t)c * KL + k0);
            br[ct] = *(const v2f*)(W_r + (size_t)c * KL + k0);
        }
#pragma unroll
        for (int t = 0; t < 4; ++t)
#pragma unroll
            for (int ct = 0; ct < 4; ++ct) {
                accl[t][ct] = wmma_f32(a[t], bl[ct], accl[t][ct]);
                accr[t][ct] = wmma_f32(a[t], br[ct], accr[t][ct]);
            }
    }
#pragma unroll
    for (int t = 0; t < 4; ++t)
#pragma unroll
        for (int ct = 0; ct < 4; ++ct) {
            const int c = ct * 16 + lid;
            const float bil = b_l[c], bir = b_r[c];
#pragma unroll
            for (int i = 0; i < 8; ++i) {
                int r = m0 + t * 16 + grp * 8 + i;        // C/D layout: M = i + 8*grp
                if (r < N) {
                    x_l[(size_t)r * CSP + c] = accl[t][ct][i] + bil;
                    x_r[(size_t)r * CSP + c] = accr[t][ct][i] + bir;
                }
            }
        }
}

// ---------------------------------------------------------------------------
// 2) e_j = att . leaky_relu(x_l[src] + x_r[dst]); segment-max via ordered-uint atomicMax
__global__ void edge_logits(const float* __restrict__ x_l, const float* __restrict__ x_r,
                            const int* __restrict__ srcv, const int* __restrict__ dstv,
                            const float* __restrict__ att,
                            float* __restrict__ e, unsigned* __restrict__ emax, int E, int N)
{
    int j = blockIdx.x * blockDim.x + threadIdx.x;
    if (j >= E + N) return;
    const int s = (j < E) ? srcv[j] : (j - E);
    const int d = (j < E) ? dstv[j] : (j - E);
    const float4* xl = (const float4*)(x_l + (size_t)s * CSP);
    const float4* xr = (const float4*)(x_r + (size_t)d * CSP);
    const float4* at = (const float4*)att;
    float acc = 0.0f;
#pragma unroll
    for (int q = 0; q < CSP / 4; ++q) {
        float4 a = xl[q], b = xr[q], w = at[q];
        acc += w.x * lrelu(a.x + b.x);
        acc += w.y * lrelu(a.y + b.y);
        acc += w.z * lrelu(a.z + b.z);
        acc += w.w * lrelu(a.w + b.w);
    }
    e[j] = acc;
    atomicMax(&emax[d], flip_f(acc));
}

// ---------------------------------------------------------------------------
// 3) e_exp = exp(e - max); segment-sum via float atomicAdd
__global__ void edge_exp(const float* __restrict__ e, const int* __restrict__ dstv,
                         const unsigned* __restrict__ emax,
                         float* __restrict__ e_exp, float* __restrict__ denom, int E, int N)
{
    int j = blockIdx.x * blockDim.x + threadIdx.x;
    if (j >= E + N) return;
    const int d = (j < E) ? dstv[j] : (j - E);
    float ex = __expf(e[j] - unflip_f(emax[d]));
    e_exp[j] = ex;
    atomicAdd(&denom[d], ex);
}

// ---------------------------------------------------------------------------
// 4) spatial[dst] += alpha * x_l[src]; thread = (edge, feature) for coalesced rows
__global__ void edge_scatter(const float* __restrict__ x_l,
                             const int* __restrict__ srcv, const int* __restrict__ dstv,
                             const float* __restrict__ e_exp, const float* __restrict__ denom,
                             float* __restrict__ spatial, int E, int N)
{
    long long tid = (long long)blockIdx.x * blockDim.x + threadIdx.x;
    if (tid >= (long long)(E + N) * CSP) return;
    const int j = (int)(tid >> 6);
    const int f = (int)(tid & (CSP - 1));
    const int s = (j < E) ? srcv[j] : (j - E);
    const int d = (j < E) ? dstv[j] : (j - E);
    const float alpha = e_exp[j] / denom[d];
    atomicAdd(&spatial[(size_t)d * CSP + f], alpha * x_l[(size_t)s * CSP + f]);
}

// ---------------------------------------------------------------------------
// 5) mean/var/sample: fused K=64 WMMA GEMMs (Wm, Wv) + exp/sqrt/rsample epilogue.
//    Same M-blocked structure: 4 row-tiles per wave, NT output stores.
__global__ __launch_bounds__(128)
void head_gemm_wmma(const float* __restrict__ spatial, const float* __restrict__ b_gat,
                    const float* __restrict__ Wm, const float* __restrict__ bm,
                    const float* __restrict__ Wv, const float* __restrict__ bv,
                    const float* __restrict__ eps,
                    float* __restrict__ out_mean, float* __restrict__ out_var,
                    float* __restrict__ out_sample, int N)
{
    const int lane = threadIdx.x & 31;
    const int wave = threadIdx.x >> 5;
    const int m0 = (blockIdx.x * 4 + wave) * 64;
    if (m0 >= N) return;
    const int grp = lane >> 4;
    const int lid = lane & 15;

    const float* sr[4];
#pragma unroll
    for (int t = 0; t < 4; ++t) {
        int r = m0 + t * 16 + lid;
        if (r >= N) r = N - 1;
        sr[t] = spatial + (size_t)r * CSP;
    }

    v8f accm[4][4] = {}, accv[4][4] = {};
    for (int k = 0; k < CSP; k += 4) {
        const int k0 = k + 2 * grp;
        const float bg0 = b_gat[k0], bg1 = b_gat[k0 + 1];
        v2f a[4], bmf[4], bvf[4];
#pragma unroll
        for (int t = 0; t < 4; ++t) {
            a[t] = *(const v2f*)(sr[t] + k0);
            a[t].x += bg0;                    // GAT output bias fused into A operand
            a[t].y += bg1;
        }
#pragma unroll
        for (int ct = 0; ct < 4; ++ct) {
            const int c = ct * 16 + lid;
            bmf[ct] = *(const v2f*)(Wm + (size_t)c * CSP + k0);
            bvf[ct] = *(const v2f*)(Wv + (size_t)c * CSP + k0);
        }
#pragma unroll
        for (int t = 0; t < 4; ++t)
#pragma unroll
            for (int ct = 0; ct < 4; ++ct) {
                accm[t][ct] = wmma_f32(a[t], bmf[ct], accm[t][ct]);
                accv[t][ct] = wmma_f32(a[t], bvf[ct], accv[t][ct]);
            }
    }
#pragma unroll
    for (int t = 0; t < 4; ++t)
#pragma unroll
        for (int ct = 0; ct < 4; ++ct) {
            const int c = ct * 16 + lid;
            const float bim = bm[c], biv = bv[c];
#pragma unroll
            for (int i = 0; i < 8; ++i) {
                int r = m0 + t * 16 + grp * 8 + i;
                if (r < N) {
                    size_t idx = (size_t)r * CSP + c;
                    float m = accm[t][ct][i] + bim;
                    float v = __expf(accv[t][ct][i] + biv) + VAR_EPS;
                    float s = m + sqrtf(v) * __builtin_nontemporal_load(eps + idx);
                    __builtin_nontemporal_store(m, out_mean + idx);
                    __builtin_nontemporal_store(v, out_var + idx);
                    __builtin_nontemporal_store(s, out_sample + idx);
                }
            }
        }
}

// ---------------------------------------------------------------------------
extern "C" void kernel_launch(void* const* d_in, const int* in_sizes, int n_in,
                              void* d_out, int out_size, void* d_ws, size_t ws_size,
                              hipStream_t stream) {
    const float* z     = (const float*)d_in[0];
    const int*   ei    = (const int*)  d_in[1];
    const float* W_l   = (const float*)d_in[2];
    const float* b_l   = (const float*)d_in[3];
    const float* W_r   = (const float*)d_in[4];
    const float* b_r   = (const float*)d_in[5];
    const float* att   = (const float*)d_in[6];
    const float* b_gat = (const float*)d_in[7];
    const float* Wm    = (const float*)d_in[8];
    const float* bm    = (const float*)d_in[9];
    const float* Wv    = (const float*)d_in[10];
    const float* bv    = (const float*)d_in[11];
    const float* eps   = (const float*)d_in[12];

    const int N = in_sizes[0] / KL;      // 50000
    const int E = in_sizes[1] / 2;       // 800000
    const int T = E + N;                 // with self loops
    const int* srcv = ei;
    const int* dstv = ei + E;

    // workspace layout
    char* ws = (char*)d_ws;
    const size_t szNC = (size_t)N * CSP * sizeof(float);
    float*    x_l     = (float*)   (ws);
    float*    x_r     = (float*)   (ws + szNC);
    float*    spatial = (float*)   (ws + 2 * szNC);
    float*    e       = (float*)   (ws + 3 * szNC);
    float*    e_exp   = (float*)   (ws + 3 * szNC + (size_t)T * sizeof(float));
    unsigned* emax    = (unsigned*)(ws + 3 * szNC + 2 * (size_t)T * sizeof(float));
    float*    denom   = (float*)   (ws + 3 * szNC + 2 * (size_t)T * sizeof(float)
                                       + (size_t)N * sizeof(unsigned));

    float* out_mean   = (float*)d_out;
    float* out_var    = out_mean + (size_t)N * CSP;
    float* out_sample = out_var  + (size_t)N * CSP;

    const int gblks  = (N + 255) / 256;          // 4 waves/block, 64 rows/wave
    const int iblks  = (N * CSP + 255) / 256;
    const int eblks  = (T + 255) / 256;
    const long long stot = (long long)T * CSP;
    const int sblks  = (int)((stot + 255) / 256);

    sp_init<<<iblks, 256, 0, stream>>>(spatial, denom, emax, N);
    xlr_gemm_wmma<<<gblks, 128, 0, stream>>>(z, W_l, b_l, W_r, b_r, x_l, x_r, N);
    edge_logits<<<eblks, 256, 0, stream>>>(x_l, x_r, srcv, dstv, att, e, emax, E, N);
    edge_exp<<<eblks, 256, 0, stream>>>(e, dstv, emax, e_exp, denom, E, N);
    edge_scatter<<<sblks, 256, 0, stream>>>(x_l, srcv, dstv, e_exp, denom, spatial, E, N);
    head_gemm_wmma<<<gblks, 128, 0, stream>>>(spatial, b_gat, Wm, bm, Wv, bv, eps,
                                              out_mean, out_var, out_sample, N);
}